// MultiplicativeLayer_13786845020927
// MI455X (gfx1250) — compile-verified
//
#include <hip/hip_runtime.h>
#include <math.h>

// MI455X / gfx1250, wave32. FP32 WMMA pipeline:
//   info = relu(info_w @ X), mask = sigmoid(mask_w @ X)
//   out  = relu(out_w @ (mask ⊗ info))
// Block = 8 waves = 128 pixels. Each wave computes info/mask for its own
// 16-pixel tile, publishes packed WMMA B-operands of E to LDS, then owns
// 2 of the 16 output-channel tiles across ALL 8 pixel tiles, so every
// out_w cacheline is read exactly once per block (256 MB total vs 2 GB
// in the naive scheme). out_w is pre-repacked into WMMA lane order in
// d_ws so A loads are contiguous 256B coalesced b64s.

#define C_IN   256
#define IC     64
#define MC     16
#define OC     256
#define EK     (MC * IC)        // 1024
#define PPB    1024             // H*W pixels per batch plane
#define KSTEPS (EK / 4)         // 256 WMMA k-steps in the out GEMM
#define CHUNK  8                // k-steps per staged E chunk (32 E rows)
#define NCHUNK (KSTEPS / CHUNK) // 32

typedef __attribute__((ext_vector_type(2))) float v2f;
typedef __attribute__((ext_vector_type(8))) float v8f;

__device__ __forceinline__ v8f wmma_f32x4(v2f a, v2f b, v8f c) {
    // V_WMMA_F32_16X16X4_F32: D = A(16x4) * B(4x16) + C(16x16)
    return __builtin_amdgcn_wmma_f32_16x16x4_f32(
        false, a, false, b, (short)0, c, false, false);
}

// Repack out_w[OC][EK] into per-(kstep, otile) WMMA A-operand lane order:
//   ws[(ks*16 + ot)*64 + lane*2 + j] =
//     out_w[(ot*16 + (lane&15))*EK + ks*4 + ((lane>>4)<<1) + j]
__global__ __launch_bounds__(256)
void repack_out_w(const float* __restrict__ out_w, float* __restrict__ ws) {
    const int tid  = blockIdx.x * 256 + threadIdx.x;   // 0..131071 v2f elems
    const int lane = tid & 31;
    const int ot   = (tid >> 5) & 15;
    const int ks   = tid >> 9;
    const v2f* src = (const v2f*)(out_w +
        (size_t)(ot * 16 + (lane & 15)) * EK + ks * 4 + ((lane >> 4) << 1));
    ((v2f*)ws)[tid] = *src;
}

__global__ __launch_bounds__(256, 1)
void mult_layer_kernel(const float* __restrict__ x,
                       const float* __restrict__ info_w,
                       const float* __restrict__ info_b,
                       const float* __restrict__ mask_w,
                       const float* __restrict__ mask_b,
                       const float* __restrict__ wsA,   // repacked out_w
                       const float* __restrict__ out_b,
                       float* __restrict__ out) {
    // Per-wave info[64][16]+mask[16][16] (40KB) + packed E chunk (16KB) = 56KB
    __shared__ float sIM[8 * 80 * 16];
    __shared__ float sE[8 * CHUNK * 64];   // [ptile][ksLocal][lane*2+j]

    const int lane  = threadIdx.x & 31;
    const int wave  = threadIdx.x >> 5;
    const int col   = lane & 15;           // N (pixel) for B/C/D; M row for A lanes
    const int khalf = (lane >> 4) << 1;    // K sub-offset: 0 or 2
    const int rowhi = (lane >> 4) << 3;    // C/D row offset: 0 or 8

    const int blkPix = blockIdx.x << 7;    // 128 pixels per block (never crosses batch)
    const int b      = blkPix >> 10;
    const int p0     = blkPix & (PPB - 1);
    const int myp    = p0 + (wave << 4);   // this wave's pixel tile (phases 1-2)

    const float* xb = x + (size_t)b * C_IN * PPB + myp;  // X[k][c] at xb[k*1024+c]
    float* sInfo = &sIM[wave * 80 * 16];                 // [64][16]
    float* sMask = sInfo + IC * 16;                      // [16][16]

    // ---------------- Phase 1: info (64x16) and mask (16x16) GEMMs ------------
    v8f accI0 = {}, accI1 = {}, accI2 = {}, accI3 = {}, accM = {};
    #pragma unroll 4
    for (int k = 0; k < C_IN; k += 4) {
        const int kk = k + khalf;
        v2f bx;                                  // B: X rows kk,kk+1 at column col
        bx.x = xb[(size_t)kk * PPB + col];
        bx.y = xb[(size_t)(kk + 1) * PPB + col];

        v2f am = *(const v2f*)(mask_w + col * C_IN + kk);
        accM = wmma_f32x4(am, bx, accM);
        v2f a0 = *(const v2f*)(info_w + (0 * 16 + col) * C_IN + kk);
        v2f a1 = *(const v2f*)(info_w + (1 * 16 + col) * C_IN + kk);
        v2f a2 = *(const v2f*)(info_w + (2 * 16 + col) * C_IN + kk);
        v2f a3 = *(const v2f*)(info_w + (3 * 16 + col) * C_IN + kk);
        accI0 = wmma_f32x4(a0, bx, accI0);
        accI1 = wmma_f32x4(a1, bx, accI1);
        accI2 = wmma_f32x4(a2, bx, accI2);
        accI3 = wmma_f32x4(a3, bx, accI3);
    }

    // Bias + activations; C/D layout: VGPR j -> row (j + rowhi), col = lane&15.
    #pragma unroll
    for (int j = 0; j < 8; ++j) {
        const int mr = j + rowhi;
        float mv = accM[j] + mask_b[mr];
        sMask[mr * 16 + col] = 1.0f / (1.0f + __expf(-mv));  // sigmoid
        sInfo[(0 * 16 + mr) * 16 + col] = fmaxf(accI0[j] + info_b[0 * 16 + mr], 0.0f);
        sInfo[(1 * 16 + mr) * 16 + col] = fmaxf(accI1[j] + info_b[1 * 16 + mr], 0.0f);
        sInfo[(2 * 16 + mr) * 16 + col] = fmaxf(accI2[j] + info_b[2 * 16 + mr], 0.0f);
        sInfo[(3 * 16 + mr) * 16 + col] = fmaxf(accI3[j] + info_b[3 * 16 + mr], 0.0f);
    }

    __syncthreads();

    // ------- Phase 3: out (256x128) = out_w(256x1024) @ E(1024x128) -----------
    // Wave w owns output tiles {2w, 2w+1} for all 8 pixel tiles.
    v8f accO[2][8];
    #pragma unroll
    for (int t = 0; t < 2; ++t)
        #pragma unroll
        for (int pt = 0; pt < 8; ++pt) accO[t][pt] = (v8f){};

    const int ot0 = wave << 1;
    const v2f* wsAv = (const v2f*)wsA;

    for (int ch = 0; ch < NCHUNK; ++ch) {
        // Build packed E B-operands for MY pixel tile. A 32-row E chunk never
        // crosses an m (mask) boundary: m = ch>>1 is chunk-invariant.
        const float mval  = sMask[(ch >> 1) * 16 + col];
        const int   ibase = (ch & 1) * 32 + khalf;        // info row base in chunk
        #pragma unroll
        for (int ksl = 0; ksl < CHUNK; ++ksl) {
            const int ir = ibase + ksl * 4;
            v2f e;
            e.x = mval * sInfo[(ir    ) * 16 + col];
            e.y = mval * sInfo[(ir + 1) * 16 + col];
            *(v2f*)&sE[(wave * CHUNK + ksl) * 64 + lane * 2] = e;
        }
        __syncthreads();

        // Consume: 2 coalesced A loads + 8 ds_load_b64 + 16 wmma per k-step.
        #pragma unroll
        for (int ksl = 0; ksl < CHUNK; ++ksl) {
            const int ks = ch * CHUNK + ksl;
            v2f a0 = wsAv[(ks * 16 + ot0    ) * 32 + lane];
            v2f a1 = wsAv[(ks * 16 + ot0 + 1) * 32 + lane];
            #pragma unroll
            for (int pt = 0; pt < 8; ++pt) {
                v2f bx = *(const v2f*)&sE[(pt * CHUNK + ksl) * 64 + lane * 2];
                accO[0][pt] = wmma_f32x4(a0, bx, accO[0][pt]);
                accO[1][pt] = wmma_f32x4(a1, bx, accO[1][pt]);
            }
        }
        __syncthreads();
    }

    // ---------------- Epilogue: bias + relu + store ---------------------------
    float* ob = out + (size_t)b * OC * PPB + p0;
    #pragma unroll
    for (int t = 0; t < 2; ++t) {
        #pragma unroll
        for (int j = 0; j < 8; ++j) {
            const int o = (ot0 + t) * 16 + j + rowhi;
            const float bias = out_b[o];
            #pragma unroll
            for (int pt = 0; pt < 8; ++pt) {
                ob[(size_t)o * PPB + pt * 16 + col] =
                    fmaxf(accO[t][pt][j] + bias, 0.0f);
            }
        }
    }
}

extern "C" void kernel_launch(void* const* d_in, const int* in_sizes, int n_in,
                              void* d_out, int out_size, void* d_ws, size_t ws_size,
                              hipStream_t stream) {
    (void)in_sizes; (void)n_in; (void)out_size; (void)ws_size;
    const float* x      = (const float*)d_in[0];
    const float* info_w = (const float*)d_in[1];
    const float* info_b = (const float*)d_in[2];
    const float* mask_w = (const float*)d_in[3];
    const float* mask_b = (const float*)d_in[4];
    const float* out_w  = (const float*)d_in[5];
    const float* out_b  = (const float*)d_in[6];
    float* out = (float*)d_out;
    float* wsA = (float*)d_ws;   // needs OC*EK*4 = 1 MB of scratch

    // Repack out_w into WMMA lane order (131072 v2f elements).
    hipLaunchKernelGGL(repack_out_w, dim3(512), dim3(256), 0, stream, out_w, wsA);

    // 32768 pixels / 128 per block = 256 blocks.
    hipLaunchKernelGGL(mult_layer_kernel, dim3(256), dim3(256), 0, stream,
                       x, info_w, info_b, mask_w, mask_b, wsA, out_b, out);
}